// InputPenmanGraph_WordEmbeddingEncoder_OutputGraph_addSelfEncodeInside_36215164240848
// MI455X (gfx1250) — compile-verified
//
#include <hip/hip_runtime.h>

// ---------------- problem constants (from reference) ----------------
constexpr int Bn  = 8;      // batch
constexpr int Tn  = 2048;   // subtokens
constexpr int Gn  = 1024;   // pooled tokens
constexpr int NN  = 600;    // node slots per sample
constexpr int NE  = 424;    // edge slots per sample
constexpr int Dn  = 1024;   // d_model
constexpr int Ln  = 2;      // graph layers
constexpr int NNB = Bn * NN;   // 4800  (= 75 * 64)
constexpr int NEB = Bn * NE;   // 3392  (= 53 * 64)

typedef float v2f __attribute__((ext_vector_type(2)));
typedef float v4f __attribute__((ext_vector_type(4)));
typedef float v8f __attribute__((ext_vector_type(8)));

// ---------------- embedding lookup + ragged mean pool ----------------
// one block per (b, g); token_group_ids are sorted per row -> binary search
__global__ __launch_bounds__(256) void embed_pool_kernel(
    const int* __restrict__ tok, const int* __restrict__ gid,
    const float* __restrict__ emb, float* __restrict__ nodes,
    float* __restrict__ edges) {
  const int b = blockIdx.x / Gn;
  const int g = blockIdx.x % Gn;
  const int* row = gid + (size_t)b * Tn;

  int lo = 0, hi = Tn;                       // lower_bound(g)
  while (lo < hi) { int mid = (lo + hi) >> 1; if (row[mid] < g) lo = mid + 1; else hi = mid; }
  const int start = lo;
  hi = Tn;                                   // lower_bound(g+1)
  while (lo < hi) { int mid = (lo + hi) >> 1; if (row[mid] < g + 1) lo = mid + 1; else hi = mid; }
  const int end = lo;

  const int cnt = end - start;
  const float inv = 1.0f / (float)(cnt > 0 ? cnt : 1);

  float* dst = (g < NN)
                   ? nodes + ((size_t)b * NN + g) * Dn
                   : edges + ((size_t)b * NE + (g - NN)) * Dn;

  const int* trow = tok + (size_t)b * Tn;
  const int d4 = threadIdx.x;  // Dn/4 == 256, one float4 per thread
  v4f s = 0.0f;
  for (int t = start; t < end; ++t) {
    const v4f* erow = (const v4f*)(emb + (size_t)trow[t] * Dn);
    s += erow[d4];
  }
  ((v4f*)dst)[d4] = s * inv;
}

// ---------------- weight pre-pack: Wp[kg][n][j] = W[4*kg+j][n] ----------------
// K-chunked transpose so a WMMA B-fragment is one contiguous float2 per lane.
__global__ __launch_bounds__(256) void pack_w_kernel(
    const float* __restrict__ W, float* __restrict__ Wp) {
  int idx = blockIdx.x * 256 + threadIdx.x;   // 0 .. (Dn/4)*Dn - 1
  int n  = idx & (Dn - 1);
  int kg = idx >> 10;
  v4f v;
  v.x = W[(size_t)(4 * kg + 0) * Dn + n];
  v.y = W[(size_t)(4 * kg + 1) * Dn + n];
  v.z = W[(size_t)(4 * kg + 2) * Dn + n];
  v.w = W[(size_t)(4 * kg + 3) * Dn + n];
  ((v4f*)Wp)[(size_t)kg * Dn + n] = v;
}

// ---------------- activation builders (float4) ----------------
__global__ __launch_bounds__(256) void build_a1_kernel(
    const float* __restrict__ nodes, const float* __restrict__ edges,
    const int* __restrict__ src, float* __restrict__ a1) {
  size_t idx = (size_t)blockIdx.x * 256 + threadIdx.x;   // over float4 elems
  if (idx >= (size_t)NEB * (Dn / 4)) return;
  int e  = (int)(idx >> 8);          // Dn/4 == 256
  int d4 = (int)(idx & 255);
  v4f n = ((const v4f*)(nodes + (size_t)src[e] * Dn))[d4];
  v4f x = ((const v4f*)edges)[idx];
  ((v4f*)a1)[idx] = n + x;
}

__global__ __launch_bounds__(256) void build_a2_kernel(
    const float* __restrict__ nodes, const int* __restrict__ src,
    const int* __restrict__ dst, float* __restrict__ a2) {
  size_t idx = (size_t)blockIdx.x * 256 + threadIdx.x;
  if (idx >= (size_t)NEB * (Dn / 4)) return;
  int e  = (int)(idx >> 8);
  int d4 = (int)(idx & 255);
  v4f n0 = ((const v4f*)(nodes + (size_t)src[e] * Dn))[d4];
  v4f n1 = ((const v4f*)(nodes + (size_t)dst[e] * Dn))[d4];
  ((v4f*)a2)[idx] = n0 + n1;
}

// ---------------- fp32 WMMA GEMM: Out[M,1024] (+)= A[M,1024] @ W[1024,1024] ----
// W is pre-packed (pack_w_kernel). block = 256 threads = 8 waves (2M x 4N);
// each wave register-blocks 32(M) x 64(N): per k-step 2 A-frags (b64) +
// 4 B-frags (b64, fully coalesced in packed layout) -> 8 independent WMMAs.
// MODE 0: Out = relu(acc)     MODE 1: Out += acc
template <int MODE>
__global__ __launch_bounds__(256) void gemm_wmma_f32_kernel(
    const float* __restrict__ A, const float* __restrict__ Wp,
    float* __restrict__ Out) {
  const int lane = threadIdx.x & 31;
  const int wv   = threadIdx.x >> 5;
  const int half = lane >> 4;      // 0 or 1
  const int hl   = lane & 15;      // 0..15
  const int wm   = wv >> 2;        // 0..1  (M direction)
  const int wn   = wv & 3;         // 0..3  (N direction)

  const int mbase = blockIdx.x * 64 + wm * 32;    // rows mbase..mbase+31
  const int nbase = blockIdx.y * 256 + wn * 64;   // cols nbase..nbase+63

  // A fragment: lane(16*half+m) holds K = 2*half + {0,1} -> contiguous float2
  const float* Arow0 = A + (size_t)(mbase + hl) * Dn + 2 * half;
  const float* Arow1 = Arow0 + (size_t)16 * Dn;
  // packed B fragment: lane(16*half+n) reads float2 at Wp[kg][n][2*half]
  const float* Bptr = Wp + (size_t)(nbase + hl) * 4 + 2 * half;

  v8f acc[2][4] = {};

#pragma unroll 2
  for (int k = 0; k < Dn; k += 4) {
    const int kg = k >> 2;
    v2f a0 = *(const v2f*)(Arow0 + k);
    v2f a1 = *(const v2f*)(Arow1 + k);
#pragma unroll
    for (int j = 0; j < 4; ++j) {
      v2f b = *(const v2f*)(Bptr + (size_t)kg * (Dn * 4) + j * 64);
      acc[0][j] = __builtin_amdgcn_wmma_f32_16x16x4_f32(
          false, a0, false, b, (short)0, acc[0][j], false, false);
      acc[1][j] = __builtin_amdgcn_wmma_f32_16x16x4_f32(
          false, a1, false, b, (short)0, acc[1][j], false, false);
    }
  }

#pragma unroll
  for (int i = 0; i < 2; ++i) {
#pragma unroll
    for (int j = 0; j < 4; ++j) {
      const int c = nbase + j * 16 + hl;
#pragma unroll
      for (int v = 0; v < 8; ++v) {
        const int r = mbase + i * 16 + v + 8 * half;  // VGPR v -> row v + 8*half
        float* p = Out + (size_t)r * Dn + c;
        float x = acc[i][j][v];
        if (MODE == 0)
          *p = fmaxf(x, 0.0f);
        else
          *p += x;
      }
    }
  }
}

// ---------------- zero + scatter-add (segment_sum over dst) ----------------
__global__ __launch_bounds__(256) void zero_kernel(float* __restrict__ p, size_t n4) {
  size_t idx = (size_t)blockIdx.x * 256 + threadIdx.x;
  if (idx < n4) ((v4f*)p)[idx] = 0.0f;
}

__global__ __launch_bounds__(256) void scatter_add_kernel(
    const float* __restrict__ msg, const int* __restrict__ dst,
    float* __restrict__ agg) {
  size_t idx = (size_t)blockIdx.x * 256 + threadIdx.x;
  if (idx >= (size_t)NEB * Dn) return;
  int e = (int)(idx >> 10);
  int d = (int)(idx & 1023);
  atomicAdd(agg + (size_t)dst[e] * Dn + d, msg[idx]);
}

// ---------------- gather back into padded [B, G, D] (float4) ----------------
__global__ __launch_bounds__(256) void copy_out_kernel(
    const float* __restrict__ nodes, const float* __restrict__ edges,
    float* __restrict__ out) {
  size_t idx = (size_t)blockIdx.x * 256 + threadIdx.x;   // over float4 elems
  if (idx >= (size_t)Bn * Gn * (Dn / 4)) return;
  int d4 = (int)(idx & 255);
  int g  = (int)((idx >> 8) & (Gn - 1));
  int b  = (int)(idx >> 18);            // G * Dn/4 == 1<<18
  v4f val = (g < NN)
                ? ((const v4f*)(nodes + ((size_t)b * NN + g) * Dn))[d4]
                : ((const v4f*)(edges + ((size_t)b * NE + (g - NN)) * Dn))[d4];
  ((v4f*)out)[idx] = val;
}

// ---------------- launch ----------------
extern "C" void kernel_launch(void* const* d_in, const int* in_sizes, int n_in,
                              void* d_out, int out_size, void* d_ws, size_t ws_size,
                              hipStream_t stream) {
  (void)in_sizes; (void)n_in; (void)out_size; (void)ws_size;

  const int*   tokens = (const int*)d_in[0];
  const int*   gids   = (const int*)d_in[1];
  /* seg_ids (d_in[2]) encodes the static nodes-first layout; unused */
  const int*   eidx   = (const int*)d_in[3];
  const float* emb    = (const float*)d_in[4];
  const float* Wmsg   = (const float*)d_in[5];
  const float* Wupd   = (const float*)d_in[6];
  const float* Wedge  = (const float*)d_in[7];

  const int* src = eidx;            // edge_index[0]
  const int* dst = eidx + NEB;      // edge_index[1]

  float* nodes = (float*)d_ws;
  float* edges = nodes + (size_t)NNB * Dn;
  float* a1    = edges + (size_t)NEB * Dn;
  float* a2    = a1    + (size_t)NEB * Dn;
  float* msg   = a2    + (size_t)NEB * Dn;
  float* agg   = msg   + (size_t)NEB * Dn;
  float* wpm   = agg   + (size_t)NNB * Dn;          // packed W_msg[l]
  float* wpe   = wpm   + (size_t)Dn * Dn;           // packed W_edge[l]
  float* wpu   = wpe   + (size_t)Dn * Dn;           // packed W_upd[l]

  const size_t nEdge4 = (size_t)NEB * (Dn / 4);
  const size_t nNode4 = (size_t)NNB * (Dn / 4);
  const int edge4Blocks = (int)((nEdge4 + 255) / 256);
  const int node4Blocks = (int)((nNode4 + 255) / 256);
  const int edgeBlocks  = (int)(((size_t)NEB * Dn + 255) / 256);
  const int packBlocks  = (Dn / 4) * Dn / 256;      // 1024

  // 1) embedding + ragged mean-pool -> nodes / edges
  embed_pool_kernel<<<Bn * Gn, 256, 0, stream>>>(tokens, gids, emb, nodes, edges);

  // 2) graph layers
  for (int l = 0; l < Ln; ++l) {
    pack_w_kernel<<<packBlocks, 256, 0, stream>>>(Wmsg  + (size_t)l * Dn * Dn, wpm);
    pack_w_kernel<<<packBlocks, 256, 0, stream>>>(Wedge + (size_t)l * Dn * Dn, wpe);
    pack_w_kernel<<<packBlocks, 256, 0, stream>>>(Wupd  + (size_t)l * Dn * Dn, wpu);

    build_a1_kernel<<<edge4Blocks, 256, 0, stream>>>(nodes, edges, src, a1);
    build_a2_kernel<<<edge4Blocks, 256, 0, stream>>>(nodes, src, dst, a2);

    // msg = relu(a1 @ Wmsg[l])
    gemm_wmma_f32_kernel<0><<<dim3(NEB / 64, Dn / 256), 256, 0, stream>>>(a1, wpm, msg);
    // edges += a2 @ Wedge[l]   (in-place; a1/a2 already captured old values)
    gemm_wmma_f32_kernel<1><<<dim3(NEB / 64, Dn / 256), 256, 0, stream>>>(a2, wpe, edges);

    // agg = segment_sum(msg, dst)
    zero_kernel<<<node4Blocks, 256, 0, stream>>>(agg, nNode4);
    scatter_add_kernel<<<edgeBlocks, 256, 0, stream>>>(msg, dst, agg);

    // nodes += agg @ Wupd[l]
    gemm_wmma_f32_kernel<1><<<dim3(NNB / 64, Dn / 256), 256, 0, stream>>>(agg, wpu, nodes);
  }

  // 3) scatter back into padded [B, G, D]
  const int out4Blocks = (int)(((size_t)Bn * Gn * (Dn / 4) + 255) / 256);
  copy_out_kernel<<<out4Blocks, 256, 0, stream>>>(nodes, edges, (float*)d_out);
}